// ExphormerAttention_18786186952962
// MI455X (gfx1250) — compile-verified
//
#include <hip/hip_runtime.h>
#include <hip/hip_bf16.h>

typedef float v2f __attribute__((ext_vector_type(2)));
typedef float v8f __attribute__((ext_vector_type(8)));

#define DIM 128   // IN_DIM == out dim
#define H 8
#define HD 16

// Accumulate C += A(16x4 f32 tile from LDS) * B(4x16 f32 tile from W), K = 0..127.
// A layout (ISA 16x4 f32): lane l holds row m=l&15, VGPR0=K(koff), VGPR1=K(koff+1),
// koff = (l>=16)?2:0.  B layout mirrors per column n.
__device__ __forceinline__ v8f wmma_gemm_tile(const float* __restrict__ xs,
                                              const float* __restrict__ W,
                                              int col, int lane, v8f c) {
    const int m    = lane & 15;
    const int koff = (lane >> 4) << 1;          // 0 or 2
    const float* ar = xs + m * DIM;
#pragma unroll
    for (int k = 0; k < DIM; k += 4) {
        v2f a, b;
        const float* ap = ar + k + koff;
        a.x = ap[0];
        a.y = ap[1];
        const float* bp = W + (size_t)(k + koff) * DIM + col;
        b.x = bp[0];
        b.y = bp[DIM];
        c = __builtin_amdgcn_wmma_f32_16x16x4_f32(false, a, false, b,
                                                  (short)0, c, false, false);
    }
    return c;
}

// ---------------- Phase 1: Q/K/V projections (x read once) ----------------
__global__ void __launch_bounds__(128)
qkv_proj_kernel(const float* __restrict__ x,
                const float* __restrict__ Wq, const float* __restrict__ bq,
                const float* __restrict__ Wk, const float* __restrict__ bk,
                const float* __restrict__ Wv, const float* __restrict__ bv,
                float* __restrict__ Q, float* __restrict__ K,
                float* __restrict__ V, int nrows) {
    __shared__ float xs[16 * DIM];              // 8 KB tile of x
    const int row0 = blockIdx.x * 16;
    const int t    = threadIdx.x;               // 128 threads = 4 waves

    // cooperative tile load (float4-coalesced); zero-fill OOB rows
    for (int i = t; i < 16 * DIM / 4; i += 128) {
        const int r = i >> 5;                   // i*4/128
        float4 v4 = make_float4(0.f, 0.f, 0.f, 0.f);
        if (row0 + r < nrows)
            v4 = ((const float4*)x)[(size_t)row0 * (DIM / 4) + i];
        ((float4*)xs)[i] = v4;
    }
    __syncthreads();

    const int wave = t >> 5;
    const int lane = t & 31;
    const float* Ws[3] = {Wq, Wk, Wv};
    const float* bs[3] = {bq, bk, bv};
    float*       Os[3] = {Q, K, V};

#pragma unroll
    for (int mtx = 0; mtx < 3; ++mtx) {
#pragma unroll
        for (int tc = 0; tc < 2; ++tc) {
            const int col  = wave * 32 + tc * 16 + (lane & 15);
            const float bb = bs[mtx][col];
            v8f c = {bb, bb, bb, bb, bb, bb, bb, bb};   // bias via C operand
            c = wmma_gemm_tile(xs, Ws[mtx], col, lane, c);
            const int rlo = (lane >> 4) << 3;           // 0 or 8
#pragma unroll
            for (int j = 0; j < 8; ++j) {
                const int row = row0 + rlo + j;
                if (row < nrows)
                    Os[mtx][(size_t)row * DIM + col] = c[j];
            }
        }
    }
}

// ------- Phase 2: fused E projection + edge attention + scatter -------
__global__ void __launch_bounds__(128)
edge_attn_kernel(const float* __restrict__ edge_attr,
                 const int* __restrict__ edge_index,   // [2, E] int32
                 const float* __restrict__ We, const float* __restrict__ be,
                 const float* __restrict__ Q, const float* __restrict__ K,
                 const float* __restrict__ V,
                 float* __restrict__ outAcc,           // wV accumulator (= d_out)
                 float* __restrict__ Z, int nEdges) {
    __shared__ float eain[16 * DIM];            // edge_attr tile
    __shared__ float eprj[16 * DIM];            // projected E tile
    const int e0 = blockIdx.x * 16;
    const int t  = threadIdx.x;

    for (int i = t; i < 16 * DIM / 4; i += 128) {
        const int r = i >> 5;
        float4 v4 = make_float4(0.f, 0.f, 0.f, 0.f);
        if (e0 + r < nEdges)
            v4 = ((const float4*)edge_attr)[(size_t)e0 * (DIM / 4) + i];
        ((float4*)eain)[i] = v4;
    }
    __syncthreads();

    const int wave = t >> 5;
    const int lane = t & 31;
#pragma unroll
    for (int tc = 0; tc < 2; ++tc) {
        const int col  = wave * 32 + tc * 16 + (lane & 15);
        const float bb = be[col];
        v8f c = {bb, bb, bb, bb, bb, bb, bb, bb};
        c = wmma_gemm_tile(eain, We, col, lane, c);
        const int rlo = (lane >> 4) << 3;
#pragma unroll
        for (int j = 0; j < 8; ++j)
            eprj[(size_t)(rlo + j) * DIM + col] = c[j];
    }
    __syncthreads();

    // 128 threads = 16 edges x 8 heads
    const int el = t >> 3;
    const int h  = t & 7;
    const int e  = e0 + el;
    if (e < nEdges) {
        const int s = edge_index[e];            // src
        const int d = edge_index[nEdges + e];   // dst
        const float4* kp = (const float4*)(K + (size_t)s * DIM + h * HD);
        const float4* qp = (const float4*)(Q + (size_t)d * DIM + h * HD);
        const float4* ep = (const float4*)(eprj + el * DIM + h * HD);
        float acc = 0.f;
#pragma unroll
        for (int i = 0; i < HD / 4; ++i) {
            const float4 kv = kp[i];
            const float4 qv = qp[i];
            const float4 ev = ep[i];
            acc += kv.x * qv.x * ev.x + kv.y * qv.y * ev.y +
                   kv.z * qv.z * ev.z + kv.w * qv.w * ev.w;
        }
        acc *= 0.25f;                           // 1/sqrt(HD)
        acc = fminf(5.f, fmaxf(-5.f, acc));
        const float sc = __expf(acc);
        atomicAdd(&Z[(size_t)d * H + h], sc);
        const float* vp = V + (size_t)s * DIM + h * HD;
        float* op = outAcc + (size_t)d * DIM + h * HD;
#pragma unroll
        for (int i = 0; i < HD; ++i)
            atomicAdd(op + i, vp[i] * sc);
    }
}

// ---------------- Phase 3: out = wV / (Z + 1e-6), in place ----------------
__global__ void __launch_bounds__(256)
finalize_kernel(float* __restrict__ out, const float* __restrict__ Z,
                size_t total) {
    const size_t i = (size_t)blockIdx.x * blockDim.x + threadIdx.x;
    if (i < total) {
        const size_t n = i >> 7;                // / DIM
        const int    h = (int)((i >> 4) & 7);   // (i % 128) / 16
        out[i] = out[i] / (Z[n * H + h] + 1e-6f);
    }
}

extern "C" void kernel_launch(void* const* d_in, const int* in_sizes, int n_in,
                              void* d_out, int out_size, void* d_ws, size_t ws_size,
                              hipStream_t stream) {
    const float* x         = (const float*)d_in[0];
    const float* edge_attr = (const float*)d_in[1];
    const int*   edge_idx  = (const int*)d_in[2];
    const float* Wq = (const float*)d_in[3];
    const float* bq = (const float*)d_in[4];
    const float* Wk = (const float*)d_in[5];
    const float* bk = (const float*)d_in[6];
    const float* We = (const float*)d_in[7];
    const float* be = (const float*)d_in[8];
    const float* Wv = (const float*)d_in[9];
    const float* bv = (const float*)d_in[10];
    float* out = (float*)d_out;

    const int N = in_sizes[0] / DIM;
    const int E = in_sizes[2] / 2;

    // workspace: Q | K | V | Z   (3*N*128 + N*8 floats)
    float* Q = (float*)d_ws;
    float* Kp = Q + (size_t)N * DIM;
    float* Vp = Kp + (size_t)N * DIM;
    float* Zp = Vp + (size_t)N * DIM;

    hipMemsetAsync(out, 0, (size_t)N * DIM * sizeof(float), stream);
    hipMemsetAsync(Zp, 0, (size_t)N * H * sizeof(float), stream);

    qkv_proj_kernel<<<(N + 15) / 16, 128, 0, stream>>>(
        x, Wq, bq, Wk, bk, Wv, bv, Q, Kp, Vp, N);

    edge_attn_kernel<<<(E + 15) / 16, 128, 0, stream>>>(
        edge_attr, edge_idx, We, be, Q, Kp, Vp, out, Zp, E);

    const size_t total = (size_t)N * DIM;
    finalize_kernel<<<(unsigned)((total + 255) / 256), 256, 0, stream>>>(
        out, Zp, total);
}